// Block_14516989461266
// MI455X (gfx1250) — compile-verified
//
#include <hip/hip_runtime.h>
#include <hip/hip_bf16.h>
#include <math.h>

// Problem constants (from reference)
#define BB      8
#define TT      64
#define DD      1024
#define UU      1024
#define MROWS   (BB*TT)          // 512 independent rows
#define TILE_M  16               // WMMA M
#define CONSENT 0.75f

// Phase-1 (ux) tiling: 32 WGs x 16 waves, 4 N-tiles per wave
#define P1_WAVES   16
#define P1_THREADS (P1_WAVES*32)
#define P1_TPW     ((UU/16)/P1_WAVES)   // 4

// Phase-2 (scan) tiling: grid (32 row-groups x 8 N-slices), 8 waves per WG,
// one 16-col N-tile per wave -> each WG owns a 16x128 slice of the output.
#define NSLICES    8
#define SLICE_N    (UU/NSLICES)          // 128
#define NGROUPS    (MROWS/TILE_M)        // 32
#define SC_WAVES   (SLICE_N/16)          // 8
#define SC_THREADS (SC_WAVES*32)         // 256

typedef float v2f __attribute__((ext_vector_type(2)));
typedef float v8f __attribute__((ext_vector_type(8)));

// ---------------------------------------------------------------------------
// Init: q0 = 0 and group-barrier counters = 0 (must be re-done every call;
// the harness does not re-poison scratch between graph replays).
// ---------------------------------------------------------------------------
__global__ void init_kernel(float* __restrict__ qbuf, unsigned* __restrict__ bar) {
    const int i = blockIdx.x * blockDim.x + threadIdx.x;
    if (i < MROWS * UU) qbuf[i] = 0.0f;
    if (i < NGROUPS)    bar[i]  = 0u;
}

// ---------------------------------------------------------------------------
// Phase 1: ux = x @ Wu + bu   ([512,1024] = [512,1024] x [1024,1024])
// ---------------------------------------------------------------------------
__global__ __launch_bounds__(P1_THREADS) void ux_gemm_kernel(
    const float* __restrict__ x, const float* __restrict__ Wu,
    const float* __restrict__ bu, float* __restrict__ ux)
{
    const int tid  = threadIdx.x;
    const int lane = tid & 31;
    const int wave = tid >> 5;
    const int g    = lane >> 4;     // half-wave group: A K-offset / C row-pair
    const int ln   = lane & 15;     // A: M ; B/C: N within tile
    const int row0 = blockIdx.x * TILE_M;

    v8f acc[P1_TPW] = {};
    for (int k = 0; k < DD; k += 4) {
        // A-frag (ISA 32-bit 16x4 layout): element r -> K = k + 2g + r
        v2f a = *(const v2f*)(x + (row0 + ln) * DD + k + 2 * g);
#pragma unroll
        for (int j = 0; j < P1_TPW; ++j) {
            const int n = (wave * P1_TPW + j) * 16 + ln;
            v2f b;
            b.x = Wu[(k + 2 * g + 0) * UU + n];
            b.y = Wu[(k + 2 * g + 1) * UU + n];
            acc[j] = __builtin_amdgcn_wmma_f32_16x16x4_f32(
                false, a, false, b, (short)0, acc[j], false, false);
        }
    }
#pragma unroll
    for (int j = 0; j < P1_TPW; ++j) {
        const int n = (wave * P1_TPW + j) * 16 + ln;
#pragma unroll
        for (int r = 0; r < 8; ++r) {
            const int m = r + 8 * g;              // C: element r -> M = r + 8g
            ux[(row0 + m) * UU + n] = acc[j][r] + bu[n];
        }
    }
}

// ---------------------------------------------------------------------------
// Group barrier: 8 slice-WGs of one row-group rendezvous on a monotonic
// agent-scope counter. Release fence before arrive, acquire fence after.
// ---------------------------------------------------------------------------
__device__ inline void group_barrier(unsigned* __restrict__ bar, unsigned target) {
    __threadfence();                 // release our global stores (agent scope)
    __syncthreads();                 // every wave of this WG has fenced
    if (threadIdx.x == 0) {
        __hip_atomic_fetch_add(bar, 1u, __ATOMIC_RELEASE, __HIP_MEMORY_SCOPE_AGENT);
        while (__hip_atomic_load(bar, __ATOMIC_ACQUIRE, __HIP_MEMORY_SCOPE_AGENT) < target) {
            __builtin_amdgcn_s_sleep(1);
        }
    }
    __syncthreads();
    __threadfence();                 // acquire: invalidate L0, see peers' stores
}

// ---------------------------------------------------------------------------
// Phase 2: 64-step latch scan. grid = (32 groups, 8 slices).
//   v  = ux + q @ Wm + bm              (q exchanged through L2 workspace)
//   gate = mean(clip(set_p*v,0,1)) >= CONSENT   (per row, slice partials)
//   vq = v @ Wv + bv                   (v exchanged through L2 workspace)
//   q  = gate ? vq : q
// ---------------------------------------------------------------------------
__global__ __launch_bounds__(SC_THREADS) void latch_scan_kernel(
    const float* __restrict__ ux, const float* __restrict__ Wm,
    const float* __restrict__ bm, const float* __restrict__ Wv,
    const float* __restrict__ bv, const float* __restrict__ set_p,
    float* __restrict__ qbuf, float* __restrict__ vbuf,
    float* __restrict__ gpart, unsigned* __restrict__ bar,
    float* __restrict__ acts, float* __restrict__ qs)
{
    __shared__ float red[TILE_M][16];
    __shared__ float gate_lds[TILE_M];

    const int tid   = threadIdx.x;
    const int lane  = tid & 31;
    const int wave  = tid >> 5;           // 0..7
    const int g     = lane >> 4;
    const int ln    = lane & 15;
    const int group = blockIdx.x;         // row tile (16 rows)
    const int slice = blockIdx.y;         // N slice (128 cols)
    const int row0  = group * TILE_M;
    const int n     = (slice * SC_WAVES + wave) * 16 + ln;
    unsigned* gbar  = bar + group;

    const float bmn = bm[n];
    const float bvn = bv[n];

    for (int step = 0; step < TT; ++step) {
        // ---------------- GEMM1 slice: c = q @ Wm[:, n] ----------------
        v8f acc = {};
        {
            const float* arow = qbuf + (row0 + ln) * UU;
            for (int k = 0; k < UU; k += 4) {
                v2f a = *(const v2f*)(arow + k + 2 * g);
                v2f b;
                b.x = Wm[(k + 2 * g + 0) * UU + n];
                b.y = Wm[(k + 2 * g + 1) * UU + n];
                acc = __builtin_amdgcn_wmma_f32_16x16x4_f32(
                    false, a, false, b, (short)0, acc, false, false);
            }
        }
        // v = c + ux + bm -> vbuf slice ; acts = tanh(v)
#pragma unroll
        for (int r = 0; r < 8; ++r) {
            const int m   = r + 8 * g;
            const int row = row0 + m;
            const float v = acc[r] + ux[row * UU + n] + bmn;
            vbuf[row * UU + n] = v;
            const int bi = row >> 6, t = row & 63;
            acts[(((bi * TT + step) * TT) + t) * UU + n] = tanhf(v);
        }
        __threadfence_block();
        __syncthreads();

        // gate partial over this WG's 128 columns (fixed-order, deterministic)
        {
            const int prow = tid >> 4;            // 0..15
            const int seg  = tid & 15;            // 0..15, 8 cols each
            const int nb   = slice * SLICE_N + seg * 8;
            float s = 0.0f;
            for (int u = 0; u < 8; ++u) {
                float sv = set_p[nb + u] * vbuf[(row0 + prow) * UU + nb + u];
                s += fminf(fmaxf(sv, 0.0f), 1.0f);
            }
            red[prow][seg] = s;
        }
        __syncthreads();
        if (tid < TILE_M) {
            float s = 0.0f;
            for (int i = 0; i < 16; ++i) s += red[tid][i];
            gpart[(group * NSLICES + slice) * TILE_M + tid] = s;
        }
        // all slices published v + gate partials; q reads of this step done
        group_barrier(gbar, (unsigned)(NSLICES * (2 * step + 1)));

        // full-row gate from all 8 slice partials (fixed order)
        if (tid < TILE_M) {
            float s = 0.0f;
            for (int sl = 0; sl < NSLICES; ++sl)
                s += gpart[(group * NSLICES + sl) * TILE_M + tid];
            gate_lds[tid] = (s * (1.0f / UU) >= CONSENT) ? 1.0f : 0.0f;
        }
        __syncthreads();

        // ---------------- GEMM2 slice: c = v @ Wv[:, n] ----------------
        acc = (v8f){};
        {
            const float* vrow = vbuf + (row0 + ln) * UU;
            for (int k = 0; k < UU; k += 4) {
                v2f a = *(const v2f*)(vrow + k + 2 * g);
                v2f b;
                b.x = Wv[(k + 2 * g + 0) * UU + n];
                b.y = Wv[(k + 2 * g + 1) * UU + n];
                acc = __builtin_amdgcn_wmma_f32_16x16x4_f32(
                    false, a, false, b, (short)0, acc, false, false);
            }
        }
        // latch update; each qbuf element is owned by exactly one lane
#pragma unroll
        for (int r = 0; r < 8; ++r) {
            const int m   = r + 8 * g;
            const int row = row0 + m;
            const float vq   = acc[r] + bvn;
            const float gt   = gate_lds[m];
            const float qold = qbuf[row * UU + n];
            const float qn   = gt * vq + (1.0f - gt) * qold;
            qbuf[row * UU + n] = qn;
            const int bi = row >> 6, t = row & 63;
            qs[(((bi * TT + step) * TT) + t) * UU + n] = qn;
        }
        // publish q for next step's GEMM1; v consumers done before overwrite
        group_barrier(gbar, (unsigned)(NSLICES * (2 * step + 2)));
    }
}

// ---------------------------------------------------------------------------
extern "C" void kernel_launch(void* const* d_in, const int* in_sizes, int n_in,
                              void* d_out, int out_size, void* d_ws, size_t ws_size,
                              hipStream_t stream) {
    (void)in_sizes; (void)n_in; (void)out_size; (void)ws_size;
    const float* x     = (const float*)d_in[0];
    const float* Wu    = (const float*)d_in[1];
    const float* bu    = (const float*)d_in[2];
    const float* Wm    = (const float*)d_in[3];
    const float* bm    = (const float*)d_in[4];
    const float* Wv    = (const float*)d_in[5];
    const float* bv    = (const float*)d_in[6];
    const float* set_p = (const float*)d_in[7];

    // Workspace layout (~6 MB): all L2-resident exchange buffers
    float*    ux    = (float*)d_ws;                      // 512*1024
    float*    qbuf  = ux    + (size_t)MROWS * UU;        // 512*1024 (latch state)
    float*    vbuf  = qbuf  + (size_t)MROWS * UU;        // 512*1024 (v exchange)
    float*    gpart = vbuf  + (size_t)MROWS * UU;        // 32*8*16 gate partials
    unsigned* bar   = (unsigned*)(gpart + NGROUPS * NSLICES * TILE_M); // 32 ctrs

    float* acts = (float*)d_out;                         // [8,64,64,1024]
    float* qs   = acts + (size_t)BB * TT * TT * UU;      // [8,64,64,1024]

    init_kernel<<<(MROWS * UU + 255) / 256, 256, 0, stream>>>(qbuf, bar);
    ux_gemm_kernel<<<dim3(NGROUPS), dim3(P1_THREADS), 0, stream>>>(x, Wu, bu, ux);
    latch_scan_kernel<<<dim3(NGROUPS, NSLICES), dim3(SC_THREADS), 0, stream>>>(
        ux, Wm, bm, Wv, bv, set_p, qbuf, vbuf, gpart, bar, acts, qs);
}